// RealGCN_75282186764763
// MI455X (gfx1250) — compile-verified
//
#include <hip/hip_runtime.h>
#include <hip/hip_bf16.h>

#define NDIM 64
#define NEG_SLOPE 0.01f

typedef __attribute__((ext_vector_type(2))) float v2f;
typedef __attribute__((ext_vector_type(8))) float v8f;

// ---------------- utility ----------------
__global__ void set_const_kernel(float* __restrict__ p, float v, int n) {
    int i = blockIdx.x * blockDim.x + threadIdx.x;
    if (i < n) p[i] = v;
}

// deg[s]++ / deg[t]++ for each original (pre-symmetrization) edge.
__global__ void degree_kernel(const long long* __restrict__ ei,
                              float* __restrict__ deg, int E) {
    int e = blockIdx.x * blockDim.x + threadIdx.x;
    if (e >= E) return;
    int s = (int)ei[e];
    int t = (int)ei[E + e];
    atomicAdd(&deg[s], 1.0f);
    atomicAdd(&deg[t], 1.0f);
}

__global__ void rsqrt_kernel(float* __restrict__ deg, int n) {
    int i = blockIdx.x * blockDim.x + threadIdx.x;
    if (i >= n) return;
    float d = deg[i];
    deg[i] = (d > 0.0f) ? rsqrtf(d) : 0.0f;
}

// x[i,:] = emb[i,:] / max(||emb[i,:]||_2, 1e-12)
__global__ void l2norm_kernel(const float* __restrict__ emb,
                              float* __restrict__ x, int n) {
    int i = blockIdx.x * blockDim.x + threadIdx.x;
    if (i >= n) return;
    const float4* row = (const float4*)(emb + (size_t)i * NDIM);
    float4* orow = (float4*)(x + (size_t)i * NDIM);
    float ss = 0.0f;
    float4 v[16];
#pragma unroll
    for (int k = 0; k < 16; ++k) {
        v[k] = row[k];
        ss += v[k].x * v[k].x + v[k].y * v[k].y + v[k].z * v[k].z + v[k].w * v[k].w;
    }
    float inv = 1.0f / fmaxf(sqrtf(ss), 1e-12f);
#pragma unroll
    for (int k = 0; k < 16; ++k) {
        float4 o;
        o.x = v[k].x * inv; o.y = v[k].y * inv;
        o.z = v[k].z * inv; o.w = v[k].w * inv;
        orow[k] = o;
    }
}

// ---------------- H = X @ W^T via V_WMMA_F32_16X16X4_F32 ----------------
// Block: 128 threads (4 waves). Block b covers node rows [16b, 16b+16).
// Wave w computes output columns [16w, 16w+16). K accumulated in 16 steps of 4.
//
// A (16x4 f32, 2 VGPRs): lane L holds row m = L&15; lanes 0-15 carry K={k0,k0+1},
// lanes 16-31 carry K={k0+2,k0+3}  (ISA 7.12.2 "32-bit A-Matrix 16x4").
// B (4x16, B[k][j] = W[j][k]): symmetric striping, lane L holds col j = L&15.
// C/D (16x16 f32, 8 VGPRs): VGPR r, lane L -> M = r + (L>=16 ? 8 : 0), Nc = L&15.
__global__ __launch_bounds__(128)
void gemm_wmma_kernel(const float* __restrict__ X, const float* __restrict__ W,
                      float* __restrict__ H, int n) {
    const int wave = threadIdx.x >> 5;
    const int lane = threadIdx.x & 31;
    const int nb = blockIdx.x * 16;         // node-row base
    const int jb = wave * 16;               // output-col base
    const int m  = lane & 15;
    const int kh = (lane >> 4) << 1;        // 0 or 2

    int rowA = nb + m;
    if (rowA >= n) rowA = n - 1;            // clamp loads; stores guarded below
    const float* arow = X + (size_t)rowA * NDIM + kh;
    const float* brow = W + (size_t)(jb + m) * NDIM + kh;  // B[k][j]=W[j][k]

    v8f c = {0.f, 0.f, 0.f, 0.f, 0.f, 0.f, 0.f, 0.f};
#pragma unroll
    for (int k0 = 0; k0 < NDIM; k0 += 4) {
        v2f a = *(const v2f*)(arow + k0);
        v2f b = *(const v2f*)(brow + k0);
        c = __builtin_amdgcn_wmma_f32_16x16x4_f32(
                /*neg_a=*/false, a, /*neg_b=*/false, b,
                /*c_mod=*/(short)0, c, /*reuse_a=*/false, /*reuse_b=*/false);
    }

    const int mhalf = (lane >> 4) * 8;
#pragma unroll
    for (int r = 0; r < 8; ++r) {
        int row = nb + mhalf + r;
        if (row < n) H[(size_t)row * NDIM + jb + m] = c[r];
    }
}

// ---------------- edge scatter: agg[t] += h[s]*norm, agg[s] += h[t]*norm ---
__global__ void scatter_kernel(const long long* __restrict__ ei,
                               const float* __restrict__ h,
                               const float* __restrict__ dinv,
                               float* __restrict__ agg, int E) {
    long long tid = (long long)blockIdx.x * blockDim.x + threadIdx.x;
    if (tid >= (long long)E * NDIM) return;
    int e = (int)(tid >> 6);
    int d = (int)(tid & 63);
    int s = (int)ei[e];
    int t = (int)ei[E + e];
    float nm = dinv[s] * dinv[t];
    atomicAdd(&agg[(size_t)t * NDIM + d], h[(size_t)s * NDIM + d] * nm);
    atomicAdd(&agg[(size_t)s * NDIM + d], h[(size_t)t * NDIM + d] * nm);
}

// ---------------- finish kernels (self-loop + bias + activation) ----------
__global__ void finish1_kernel(const float* __restrict__ agg, const float* __restrict__ h,
                               const float* __restrict__ dinv, const float* __restrict__ b,
                               float* __restrict__ x1, int nf) {
    int i = blockIdx.x * blockDim.x + threadIdx.x;
    if (i >= nf) return;
    int node = i >> 6, d = i & 63;
    float di = dinv[node];
    float v = agg[i] + di * di * h[i] + b[d];
    x1[i] = (v > 0.0f) ? v : NEG_SLOPE * v;
}

__global__ void finish2_kernel(const float* __restrict__ agg, const float* __restrict__ h,
                               const float* __restrict__ dinv, const float* __restrict__ b,
                               const float* __restrict__ x, const float* __restrict__ x1,
                               float* __restrict__ out, int nf) {
    int i = blockIdx.x * blockDim.x + threadIdx.x;
    if (i >= nf) return;
    int node = i >> 6, d = i & 63;
    float di = dinv[node];
    float v = agg[i] + di * di * h[i] + b[d];
    v = (v > 0.0f) ? v : NEG_SLOPE * v;          // x2
    out[i] = x[i] + x1[i] + v;                   // partial sum (x3 added later)
}

__global__ void finish3_kernel(const float* __restrict__ agg, const float* __restrict__ h,
                               const float* __restrict__ dinv, const float* __restrict__ b,
                               float* __restrict__ out, int nf) {
    int i = blockIdx.x * blockDim.x + threadIdx.x;
    if (i >= nf) return;
    int node = i >> 6, d = i & 63;
    float di = dinv[node];
    out[i] += agg[i] + di * di * h[i] + b[d];    // layer 3: no activation
}

// ---------------- driver ----------------
extern "C" void kernel_launch(void* const* d_in, const int* in_sizes, int n_in,
                              void* d_out, int out_size, void* d_ws, size_t ws_size,
                              hipStream_t stream) {
    const long long* ei = (const long long*)d_in[0];
    const float* emb = (const float*)d_in[1];
    const float* W1 = (const float*)d_in[2];
    const float* b1 = (const float*)d_in[3];
    const float* W2 = (const float*)d_in[4];
    const float* b2 = (const float*)d_in[5];
    const float* W3 = (const float*)d_in[6];
    const float* b3 = (const float*)d_in[7];
    float* out = (float*)d_out;

    const int E  = in_sizes[0] / 2;
    const int N  = in_sizes[1] / NDIM;
    const int NF = N * NDIM;

    float* ws   = (float*)d_ws;
    float* dinv = ws;            // N      (holds deg, then rsqrt in place)
    float* x    = ws + N;        // N*64   normalized embeddings
    float* x1   = x + NF;        // N*64
    float* h    = x1 + NF;       // N*64   per-layer GEMM output
    float* agg  = h + NF;        // N*64   per-layer scatter accumulator

    const int T = 256;
    dim3 gemmGrid((N + 15) / 16), gemmBlock(128);
    long long totEd = (long long)E * NDIM;
    int scatBlocks = (int)((totEd + T - 1) / T);

    // graph normalization
    set_const_kernel<<<(N + T - 1) / T, T, 0, stream>>>(dinv, 1.0f, N); // self-loop
    degree_kernel<<<(E + T - 1) / T, T, 0, stream>>>(ei, dinv, E);
    rsqrt_kernel<<<(N + T - 1) / T, T, 0, stream>>>(dinv, N);
    l2norm_kernel<<<(N + T - 1) / T, T, 0, stream>>>(emb, x, N);

    // layer 1: x1 = lrelu(conv(x, W1, b1))
    gemm_wmma_kernel<<<gemmGrid, gemmBlock, 0, stream>>>(x, W1, h, N);
    set_const_kernel<<<(NF + T - 1) / T, T, 0, stream>>>(agg, 0.0f, NF);
    scatter_kernel<<<scatBlocks, T, 0, stream>>>(ei, h, dinv, agg, E);
    finish1_kernel<<<(NF + T - 1) / T, T, 0, stream>>>(agg, h, dinv, b1, x1, NF);

    // layer 2: out = x + x1 + lrelu(conv(x1, W2, b2))
    gemm_wmma_kernel<<<gemmGrid, gemmBlock, 0, stream>>>(x1, W2, h, N);
    set_const_kernel<<<(NF + T - 1) / T, T, 0, stream>>>(agg, 0.0f, NF);
    scatter_kernel<<<scatBlocks, T, 0, stream>>>(ei, h, dinv, agg, E);
    finish2_kernel<<<(NF + T - 1) / T, T, 0, stream>>>(agg, h, dinv, b2, x, x1, out, NF);

    // layer 3: out += conv(x1, W3, b3)   (no activation)
    gemm_wmma_kernel<<<gemmGrid, gemmBlock, 0, stream>>>(x1, W3, h, N);
    set_const_kernel<<<(NF + T - 1) / T, T, 0, stream>>>(agg, 0.0f, NF);
    scatter_kernel<<<scatBlocks, T, 0, stream>>>(ei, h, dinv, agg, E);
    finish3_kernel<<<(NF + T - 1) / T, T, 0, stream>>>(agg, h, dinv, b3, out, NF);
}